// LCOT_torch_36532991820579
// MI455X (gfx1250) — compile-verified
//
#include <hip/hip_runtime.h>
#include <math.h>

#define ROW_N   4096            // samples per row (fixed by reference)
#define REF_N   1024            // REFSIZE
#define XN      (3 * REF_N)     // 3072 extended-CDF points
#define BLOCK   256             // 8 wave32 per row
#define EPS32   1.1920928955078125e-07f

typedef __attribute__((ext_vector_type(2))) float v2f;
typedef __attribute__((ext_vector_type(8))) float v8f;

#define USE_ASYNC 1
#define HAS_WMMA_F32X4 __has_builtin(__builtin_amdgcn_wmma_f32_16x16x4_f32)

// Flat address of a __shared__ object: low 32 bits == LDS byte offset
// (ISA 10.2: LDS aperture base has zero low 32 bits).
__device__ __forceinline__ unsigned lds_byte_off(const void* p) {
  return (unsigned)(unsigned long long)(size_t)p;
}

__global__ __launch_bounds__(BLOCK) void lcot_row_kernel(
    const float* __restrict__ x1, const float* __restrict__ w1,
    float* __restrict__ rowOut)
{
  __shared__ __align__(16) float xsL[ROW_N];   // samples (sorted in place)
  __shared__ __align__(16) float wsL[ROW_N];   // weights -> inclusive cumsum (ys)
  __shared__ __align__(16) float ecdf[XN];     // extended CDF
  __shared__ float chunkS[BLOCK];              // 256 chunk totals (16 elems/chunk)
  __shared__ float redA[BLOCK];
  __shared__ float redB[BLOCK];
  __shared__ float alphaSh;

  const int tid = threadIdx.x;
  const int row = blockIdx.x;
  const float* __restrict__ xr = x1 + (size_t)row * ROW_N;
  const float* __restrict__ wr = w1 + (size_t)row * ROW_N;

  // ---- Phase 1: stage row into LDS (async tensor path: global -> LDS) ----
#if USE_ASYNC
  #pragma unroll
  for (int k = 0; k < 4; ++k) {
    const int e = (k * BLOCK + tid) * 4;   // 16B per lane, coalesced
    asm volatile("global_load_async_to_lds_b128 %0, %1, off"
                 :: "v"(lds_byte_off(&xsL[e])),
                    "v"((unsigned long long)(const void*)(xr + e))
                 : "memory");
    asm volatile("global_load_async_to_lds_b128 %0, %1, off"
                 :: "v"(lds_byte_off(&wsL[e])),
                    "v"((unsigned long long)(const void*)(wr + e))
                 : "memory");
  }
  asm volatile("s_wait_asynccnt 0" ::: "memory");
#else
  #pragma unroll
  for (int k = 0; k < 4; ++k) {
    const int e = (k * BLOCK + tid) * 4;
    const float4 xv = *(const float4*)(xr + e);
    const float4 wv = *(const float4*)(wr + e);
    *(float4*)&xsL[e] = xv;
    *(float4*)&wsL[e] = wv;
  }
#endif
  __syncthreads();

  // ---- Phase 2: alpha = sum(x*w)/sum(w) - 0.5 (order-invariant: do pre-sort) ----
  {
    float sxw = 0.f, sw = 0.f;
    for (int i = tid; i < ROW_N; i += BLOCK) {
      const float x = xsL[i], w = wsL[i];
      sxw += x * w; sw += w;
    }
    redA[tid] = sxw; redB[tid] = sw;
    __syncthreads();
    for (int d = BLOCK >> 1; d > 0; d >>= 1) {
      if (tid < d) { redA[tid] += redA[tid + d]; redB[tid] += redB[tid + d]; }
      __syncthreads();
    }
    if (tid == 0) alphaSh = redA[0] / redB[0] - 0.5f;
  }

  // ---- Phase 3: bitonic sort of (key=xsL, payload=wsL), 4096 elems ----
  for (int k = 2; k <= ROW_N; k <<= 1) {
    for (int j = k >> 1; j > 0; j >>= 1) {
      __syncthreads();
      for (int t = tid; t < ROW_N / 2; t += BLOCK) {
        const int i = ((t & ~(j - 1)) << 1) | (t & (j - 1));
        const int p = i | j;
        const bool up = ((i & k) == 0);
        const float ka = xsL[i], kb = xsL[p];
        if ((ka > kb) == up) {
          xsL[i] = kb; xsL[p] = ka;
          const float wa = wsL[i], wb = wsL[p];
          wsL[i] = wb; wsL[p] = wa;
        }
      }
    }
  }
  __syncthreads();

  // ---- Phase 4: inclusive cumsum of wsL -> ys (in place) ----
  // Intra-chunk (16-elem) scans as a matmul: D = L(16x16 lower-tri ones) x B,
  // chained V_WMMA_F32_16X16X4_F32 over four K-slices. One wave per 256-elem
  // tile; uniform trip count (2 tiles/wave) so EXEC stays all-ones around the
  // WMMA chain with no divergent-loop exec-mask bookkeeping.
#if HAS_WMMA_F32X4
  {
    const int lane = tid & 31;
    const int wv   = tid >> 5;
    const bool hi  = lane >= 16;
    const int  l16 = hi ? lane - 16 : lane;
    #pragma unroll
    for (int it = 0; it < (ROW_N / 256) / (BLOCK / 32); ++it) {  // 2 iters
      const int t    = wv + it * (BLOCK / 32);
      const int base = t * 256;
      v8f c = {};
      #pragma unroll
      for (int s = 0; s < 4; ++s) {
        // A (16x4 f32): VGPR0 holds K=0 (lanes 0-15) / K=2 (lanes 16-31),
        //               VGPR1 holds K=1 / K=3.  A[m][k] = (k <= m).
        const int k0 = 4 * s + (hi ? 2 : 0);
        v2f a, b;
        a.x = (k0     <= l16) ? 1.0f : 0.0f;
        a.y = (k0 + 1 <= l16) ? 1.0f : 0.0f;
        // B (4x16 f32), layout symmetric to A: B[k][n] = w[base + n*16 + k]
        b.x = wsL[base + l16 * 16 + k0];
        b.y = wsL[base + l16 * 16 + k0 + 1];
        c = __builtin_amdgcn_wmma_f32_16x16x4_f32(false, a, false, b,
                                                  (short)0, c, false, false);
      }
      // D (16x16 f32): VGPR v -> M=v (lanes 0-15), M=v+8 (lanes 16-31); N=lane%16.
      // D[m][n] = inclusive scan of chunk n at element m.
      #pragma unroll
      for (int v = 0; v < 8; ++v) {
        const int m = hi ? (v + 8) : v;
        wsL[base + l16 * 16 + m] = c[v];
      }
      if (hi) chunkS[t * 16 + l16] = c[7];   // D[15][n] = chunk total
    }
  }
#else
  {
    float run = 0.f;
    const int b0 = tid * 16;
    #pragma unroll
    for (int i = 0; i < 16; ++i) { run += wsL[b0 + i]; wsL[b0 + i] = run; }
    chunkS[tid] = run;
  }
#endif
  __syncthreads();
  // Hillis-Steele inclusive scan of the 256 chunk totals.
  for (int d = 1; d < BLOCK; d <<= 1) {
    const float v = chunkS[tid];
    const float u = (tid >= d) ? chunkS[tid - d] : 0.f;
    __syncthreads();
    chunkS[tid] = v + u;
    __syncthreads();
  }
  {
    const float off = (tid == 0) ? 0.f : chunkS[tid - 1];
    const int b0 = tid * 16;
    #pragma unroll
    for (int i = 0; i < 16; ++i) wsL[b0 + i] += off;
  }
  __syncthreads();

  // ---- Phase 5: extended CDF: ecdf[i] = floor(xnew) + interp1d(xs, ys, frac(xnew)) ----
  const float* __restrict__ ys = wsL;
  for (int i = tid; i < XN; i += BLOCK) {
    const float xn = -1.0f + (3.0f * (float)i) / (float)(XN - 1);
    const float ix = floorf(xn);
    const float rx = xn - ix;
    int lo = 0, hib = ROW_N;                 // searchsorted side='left'
    while (lo < hib) { const int mid = (lo + hib) >> 1;
                       if (xsL[mid] < rx) lo = mid + 1; else hib = mid; }
    int ind = lo - 1;
    if (ind < 0) ind = 0; if (ind > ROW_N - 2) ind = ROW_N - 2;
    const float x0 = xsL[ind], x1v = xsL[ind + 1];
    const float y0 = ys[ind],  y1v = ys[ind + 1];
    const float s  = (y1v - y0) / (EPS32 + (x1v - x0));
    ecdf[i] = ix + y0 + s * (rx - x0);
  }
  __syncthreads();

  // ---- Phase 6: inverse CDF embedding + circular cost, per-row sum ----
  {
    const float alpha = alphaSh;
    float acc = 0.f;
    for (int g = tid; g < REF_N; g += BLOCK) {
      const float gx = (float)g / (float)REF_N;
      const float q  = gx - alpha;
      int lo = 0, hib = XN;
      while (lo < hib) { const int mid = (lo + hib) >> 1;
                         if (ecdf[mid] < q) lo = mid + 1; else hib = mid; }
      int ind = lo - 1;
      if (ind < 0) ind = 0; if (ind > XN - 2) ind = XN - 2;
      const float e0  = ecdf[ind], e1 = ecdf[ind + 1];
      const float xn0 = -1.0f + (3.0f * (float)ind) / (float)(XN - 1);
      const float dxn = 3.0f / (float)(XN - 1);
      const float s   = dxn / (EPS32 + (e1 - e0));
      const float emb = xn0 + s * (q - e0) - gx;
      const float a   = fabsf(emb);
      const float m   = fminf(a, 1.0f - a);
      acc += m * m;
    }
    redA[tid] = acc;
    __syncthreads();
    for (int d = BLOCK >> 1; d > 0; d >>= 1) {
      if (tid < d) redA[tid] += redA[tid + d];
      __syncthreads();
    }
    if (tid == 0) rowOut[row] = redA[0];
  }
}

__global__ __launch_bounds__(BLOCK) void lcot_finalize_kernel(
    const float* __restrict__ rowSums, float* __restrict__ out, int l)
{
  __shared__ float red[BLOCK];
  float s = 0.f;
  for (int i = threadIdx.x; i < l; i += BLOCK) s += rowSums[i];
  red[threadIdx.x] = s;
  __syncthreads();
  for (int d = BLOCK >> 1; d > 0; d >>= 1) {
    if (threadIdx.x < d) red[threadIdx.x] += red[threadIdx.x + d];
    __syncthreads();
  }
  if (threadIdx.x == 0) out[0] = sqrtf(red[0] / (float)l + 1e-8f);
}

extern "C" void kernel_launch(void* const* d_in, const int* in_sizes, int n_in,
                              void* d_out, int out_size, void* d_ws, size_t ws_size,
                              hipStream_t stream) {
  (void)n_in; (void)out_size; (void)ws_size;
  const float* x1 = (const float*)d_in[0];
  const float* w1 = (const float*)d_in[1];
  float* out      = (float*)d_out;
  const int l     = in_sizes[0] / ROW_N;      // 2048
  float* rowSums  = (float*)d_ws;             // l floats of scratch

  lcot_row_kernel<<<l, BLOCK, 0, stream>>>(x1, w1, rowSums);
  lcot_finalize_kernel<<<1, BLOCK, 0, stream>>>(rowSums, out, l);
}